// GNSMsg_38989713113819
// MI455X (gfx1250) — compile-verified
//
#include <hip/hip_runtime.h>
#include <math.h>

#define NB 4
#define NN 512
#define NODES (NB * NN)
#define DD 32
#define IND 68          // 4 + 2*D
#define LDF 80          // padded leading dim for feats / hidden
#define KITER 30

typedef __attribute__((ext_vector_type(2))) float v2f;
typedef __attribute__((ext_vector_type(8))) float v8f;
typedef unsigned int v4u __attribute__((ext_vector_type(4)));
typedef int          v4i __attribute__((ext_vector_type(4)));
typedef int          v8i __attribute__((ext_vector_type(8)));

#if defined(__has_builtin)
#if __has_builtin(__builtin_amdgcn_tensor_load_to_lds) && __has_builtin(__builtin_amdgcn_s_wait_tensorcnt)
#define HAVE_TDM 1
#endif
#endif

__device__ __forceinline__ float wave_sum(float x) {
    #pragma unroll
    for (int off = 16; off > 0; off >>= 1) x += __shfl_xor(x, off, 32);
    return x;
}

#ifdef HAVE_TDM
// TDM 2D tile load: copy K rows x Npad cols (f32) of B[K x ldb] into LDS at
// byte offset lds_off; cols >= N read zeros (TDM OOB semantics) -> free padding.
// Descriptor packing per CDNA5 ISA sec. 8.3/8.4.
__device__ __forceinline__ void tdm_load_b(const float* gptr, int K, int N,
                                           int ldb, int Npad, unsigned lds_off) {
    unsigned long long ga = (unsigned long long)(size_t)gptr;
    v4u g0;
    g0[0] = 1u;                                           // count=1 (valid user D#)
    g0[1] = lds_off;                                      // lds_addr (bytes)
    g0[2] = (unsigned)(ga & 0xFFFFFFFFu);                 // global_addr[31:0]
    g0[3] = (unsigned)((ga >> 32) & 0x01FFFFFFu) | (2u << 30);  // addr[56:32] | type=2
    unsigned td0 = (unsigned)N;         // tensor_dim0: OOB cols return 0
    unsigned td1 = (unsigned)K;
    unsigned t0  = (unsigned)Npad;      // tile_dim0 (LDS row pitch)
    unsigned t1  = (unsigned)K;        // tile_dim1
    unsigned long long s0 = (unsigned long long)(unsigned)ldb;
    v8i g1;
    g1[0] = (int)(2u << 16);                              // wg_mask=0, data_size=4B
    g1[1] = (int)((td0 & 0xFFFFu) << 16);
    g1[2] = (int)(((td0 >> 16) & 0xFFFFu) | ((td1 & 0xFFFFu) << 16));
    g1[3] = (int)(((td1 >> 16) & 0xFFFFu) | ((t0 & 0xFFFFu) << 16));
    g1[4] = (int)(t1 & 0xFFFFu);
    g1[5] = (int)(s0 & 0xFFFFFFFFu);
    g1[6] = (int)((s0 >> 32) & 0xFFFFu);
    g1[7] = 0;
    v4i gz = {0, 0, 0, 0};
#if defined(__clang_major__) && (__clang_major__ >= 23)
    v8i gz8 = {0, 0, 0, 0, 0, 0, 0, 0};
    __builtin_amdgcn_tensor_load_to_lds(g0, g1, gz, gz, gz8, 0);
#else
    __builtin_amdgcn_tensor_load_to_lds(g0, g1, gz, gz, 0);
#endif
}
#endif

// ---------------------------------------------------------------- init
__global__ void k_init(const float* __restrict__ V0,
                       const float* __restrict__ Yr, const float* __restrict__ Yi,
                       float* __restrict__ v, float* __restrict__ th,
                       float* __restrict__ m, float* __restrict__ rs,
                       float* __restrict__ sumA, float* __restrict__ loss) {
    int tid  = threadIdx.x;
    int lane = tid & 31;
    int node = blockIdx.x * 8 + (tid >> 5);
    if (blockIdx.x == 0 && tid == 0) loss[0] = 0.f;
    if (node >= NODES) return;
    int i = node % NN;
    const float* yr_row = Yr + (size_t)node * NN;
    const float* yi_row = Yi + (size_t)node * NN;
    float cnt = 0.f;
    for (int j = lane; j < NN; j += 32) {
        float yr = yr_row[j], yi = yi_row[j];
        if ((j != i) && (yr != 0.f || yi != 0.f)) cnt += 1.f;
    }
    cnt = wave_sum(cnt);
    m[node * DD + lane] = 0.f;
    if (lane == 0) {
        v[node]  = V0[node * 2 + 0];
        th[node] = V0[node * 2 + 1];
        float rc = fmaxf(cnt, 1.f);
        rs[node]   = rc;
        sumA[node] = cnt / rc;
    }
}

// ---------------------------------------------------------------- power flow
__global__ void k_powerflow(const float* __restrict__ Yr, const float* __restrict__ Yi,
                            const int* __restrict__ bus_type,
                            const float* __restrict__ Pset, const float* __restrict__ Qset,
                            const float* __restrict__ v, const float* __restrict__ th,
                            float* __restrict__ dP, float* __restrict__ dQ,
                            float* __restrict__ loss, float wk) {
    __shared__ float sc[NN], ss[NN];
    int tid = threadIdx.x;
    int b   = blockIdx.x / (NN / 8);
    int rb  = blockIdx.x % (NN / 8);
    for (int j = tid; j < NN; j += blockDim.x) {
        float vv = v[b * NN + j], tt = th[b * NN + j];
        sc[j] = vv * cosf(tt);
        ss[j] = vv * sinf(tt);
    }
    __syncthreads();
    int lane = tid & 31;
    int i    = rb * 8 + (tid >> 5);
    int node = b * NN + i;
    const float* yr_row = Yr + (size_t)node * NN;
    const float* yi_row = Yi + (size_t)node * NN;
    float ir = 0.f, ii = 0.f;
    for (int j = lane; j < NN; j += 32) {
        float yr = yr_row[j], yi = yi_row[j];
        float c = sc[j], s = ss[j];
        ir = fmaf(yr, c, fmaf(-yi, s, ir));
        ii = fmaf(yr, s, fmaf( yi, c, ii));
    }
    ir = wave_sum(ir);
    ii = wave_sum(ii);
    if (lane == 0) {
        float c = sc[i], s = ss[i];
        float P = c * ir + s * ii;
        float Q = s * ir - c * ii;
        int bt = bus_type[node];
        bool slack = (bt == 1);
        bool pvs   = slack || (bt == 2);
        float dp = slack ? 0.f : (Pset[node] - P);
        float dq = pvs   ? 0.f : (Qset[node] - Q);
        dP[node] = dp;
        dQ[node] = dq;
        atomicAdd(loss, wk * (dp * dp + dq * dq) * (1.0f / (float)NODES));
    }
}

// ---------------------------------------------------------------- WMMA GEMM
// C[M,N] = act(A[M,K] @ B[K,N] + bias) in f32 via V_WMMA_F32_16X16X4_F32.
// KK and NPAD are compile-time so the fully-unrolled K-loop uses immediate
// DS/global offsets (no per-load address VALU).  B is staged into LDS (TDM
// when available) zero-padded to NPAD.  One wave per 16x16 tile; the tile
// guard is wave-uniform so EXEC stays all-ones around every WMMA.
// act: 0 = none, 1 = leaky_relu(0.01), 2 = tanh
template<int KK, int NPAD>
__global__ __launch_bounds__(256)
void k_wmma_gemm(const float* __restrict__ A, const float* __restrict__ Bm,
                 const float* __restrict__ bias, float* __restrict__ C,
                 int M, int N, int lda, int ldb, int ldc, int act) {
    __shared__ float sB[KK * NPAD];
    int tid  = threadIdx.x;
    int lane = tid & 31;
    constexpr int tilesN = NPAD >> 4;
    int tilesM = M >> 4;

#ifdef HAVE_TDM
    if (tid < 32) {            // one wave issues the DMA; EXEC ignored by TDM
        tdm_load_b(Bm, KK, N, ldb, NPAD, (unsigned)(size_t)(&sB[0]));
        __builtin_amdgcn_s_wait_tensorcnt(0);
    }
#else
    for (int idx = tid; idx < KK * NPAD; idx += 256) {
        int kk = idx / NPAD, c = idx - kk * NPAD;
        sB[idx] = (c < N) ? Bm[kk * ldb + c] : 0.f;
    }
#endif
    __syncthreads();

    int tile = blockIdx.x * 8 + (tid >> 5);
    if (tile < tilesM * tilesN) {            // wave-uniform guard
        int tm = tile / tilesN, tn = tile % tilesN;
        int row0 = tm << 4, col0 = tn << 4;
        int hf = lane >> 4, l16 = lane & 15;
        int ar = row0 + l16;
        int bc = col0 + l16;
        const float* arow = A + (size_t)ar * lda + 2 * hf;   // loop-invariant base
        const float* brow = &sB[2 * hf * NPAD + bc];          // loop-invariant base
        v8f acc = {};
        #pragma unroll
        for (int k = 0; k < KK; k += 4) {
            v2f a = *(const v2f*)(arow + k);     // b64, immediate offset k*4
            v2f b;
            b.x = brow[k * NPAD];                // ds_load, immediate offset
            b.y = brow[(k + 1) * NPAD];
            acc = __builtin_amdgcn_wmma_f32_16x16x4_f32(
                      false, a, false, b, (short)0, acc, false, false);
        }
        if (bc < N) {
            float bv = bias ? bias[bc] : 0.f;
            #pragma unroll
            for (int r = 0; r < 8; ++r) {
                int rr  = row0 + r + 8 * hf;
                float x = acc[r] + bv;
                if (act == 1)      x = (x >= 0.f) ? x : 0.01f * x;
                else if (act == 2) x = tanhf(x);
                C[rr * ldc + bc] = x;
            }
        }
    }
}

// ---------------------------------------------------------------- edge reduce
__global__ void k_edge(const float* __restrict__ Yr, const float* __restrict__ Yi,
                       const float* __restrict__ ew1, const float* __restrict__ ew2,
                       const float* __restrict__ eb2,
                       const float* __restrict__ base, const float* __restrict__ rs,
                       const float* __restrict__ sumA, const float* __restrict__ m,
                       const float* __restrict__ v, const float* __restrict__ th,
                       const float* __restrict__ dP, const float* __restrict__ dQ,
                       float* __restrict__ Mmsg, float* __restrict__ feats) {
    __shared__ float sH[8][DD];
    __shared__ float sHH[DD];
    int node = blockIdx.x;
    int i    = node % NN;
    int tid  = threadIdx.x;
    int jg = tid >> 5, d = tid & 31;
    float base_d = base[node * DD + d];
    float w1r = ew1[DD * DD + d];           // row 32 of edge_w1 (Yr coeff)
    float w1i = ew1[DD * DD + DD + d];      // row 33 of edge_w1 (Yi coeff)
    const float* yr_row = Yr + (size_t)node * NN;
    const float* yi_row = Yi + (size_t)node * NN;
    float acc = 0.f;
    for (int j = jg; j < NN; j += 8) {
        float yr = yr_row[j], yi = yi_row[j];
        if ((j != i) && (yr != 0.f || yi != 0.f)) {
            float pre = fmaf(yr, w1r, fmaf(yi, w1i, base_d));
            acc += fmaxf(pre, 0.f);
        }
    }
    sH[jg][d] = acc;
    __syncthreads();
    if (tid < DD) {
        float H = 0.f;
        #pragma unroll
        for (int g = 0; g < 8; ++g) H += sH[g][d];
        sHH[d] = H / rs[node];
    }
    __syncthreads();
    if (tid < DD) {
        float msg = sumA[node] * eb2[d];
        #pragma unroll 8
        for (int e = 0; e < DD; ++e) msg = fmaf(sHH[e], ew2[e * DD + d], msg);
        Mmsg[node * DD + d] = msg;
        float* fr = feats + (size_t)node * LDF;
        fr[4 + d]      = m[node * DD + d];
        fr[4 + DD + d] = msg;
        if (d < 4) {
            float val = (d == 0) ? v[node] : (d == 1) ? th[node]
                      : (d == 2) ? dP[node] : dQ[node];
            fr[d] = val;
        }
        if (d < LDF - IND) fr[IND + d] = 0.f;   // zero pad cols 68..79
    }
}

// ---------------------------------------------------------------- out=1 head
__global__ void k_outdot(const float* __restrict__ h2, const float* __restrict__ w3,
                         const float* __restrict__ b3, float* __restrict__ out) {
    int tid  = threadIdx.x;
    int lane = tid & 31;
    int row  = blockIdx.x * 8 + (tid >> 5);
    const float* hr = h2 + (size_t)row * LDF;
    float s = 0.f;
    for (int e = lane; e < IND; e += 32) s = fmaf(hr[e], w3[e], s);
    s = wave_sum(s);
    if (lane == 0) out[row] = s + b3[0];
}

// ---------------------------------------------------------------- update
__global__ void k_update(const int* __restrict__ bus_type,
                         const float* __restrict__ dth, const float* __restrict__ dv,
                         const float* __restrict__ dm,
                         float* __restrict__ v, float* __restrict__ th,
                         float* __restrict__ m) {
    int idx = blockIdx.x * blockDim.x + threadIdx.x;
    if (idx >= NODES * DD) return;
    int node = idx >> 5, d = idx & 31;
    m[idx] += dm[idx];
    if (d == 0) {
        int bt = bus_type[node];
        bool slack = (bt == 1);
        bool pvs   = slack || (bt == 2);
        if (!slack) th[node] += dth[node];
        if (!pvs)   v[node]  += dv[node];
    }
}

// ---------------------------------------------------------------- output pack
__global__ void k_final(const float* __restrict__ v, const float* __restrict__ th,
                        const float* __restrict__ loss, float* __restrict__ out) {
    int idx = blockIdx.x * blockDim.x + threadIdx.x;
    if (idx < NODES) {
        out[idx * 2 + 0] = v[idx];
        out[idx * 2 + 1] = th[idx];
    } else if (idx == NODES) {
        out[NODES * 2] = loss[0];
    }
}

// ================================================================ host
extern "C" void kernel_launch(void* const* d_in, const int* in_sizes, int n_in,
                              void* d_out, int out_size, void* d_ws, size_t ws_size,
                              hipStream_t stream) {
    (void)in_sizes; (void)n_in; (void)out_size; (void)ws_size;

    const int*   bus_type = (const int*)  d_in[0];
    const float* Yr       = (const float*)d_in[1];
    const float* Yi       = (const float*)d_in[2];
    const float* P_set    = (const float*)d_in[3];
    const float* Q_set    = (const float*)d_in[4];
    const float* V0       = (const float*)d_in[5];
    const float* edge_w1  = (const float*)d_in[6];
    const float* edge_b1  = (const float*)d_in[7];
    const float* edge_w2  = (const float*)d_in[8];
    const float* edge_b2  = (const float*)d_in[9];
    const float* W[3][6];
    for (int p = 0; p < 3; ++p)
        for (int q = 0; q < 6; ++q)
            W[p][q] = (const float*)d_in[10 + p * 6 + q];

    float* ws   = (float*)d_ws;
    float* loss = ws;                 // 1 (block of 16 for alignment)
    float* v    = ws + 16;            // 2048
    float* th   = v    + NODES;
    float* rs   = th   + NODES;
    float* sumA = rs   + NODES;
    float* dP   = sumA + NODES;
    float* dQ   = dP   + NODES;
    float* dv   = dQ   + NODES;
    float* dth  = dv   + NODES;
    float* m    = dth  + NODES;       // 2048*32
    float* base = m    + NODES * DD;
    float* Mmsg = base + NODES * DD;
    float* dm   = Mmsg + NODES * DD;
    float* feats= dm   + NODES * DD;  // 2048*80
    float* h1   = feats+ NODES * LDF;
    float* h2   = h1   + NODES * LDF;

    // tile counts: M=2048 -> 128 row tiles; blocks = tiles/8 waves
    const int blk3232 = (128 * (32 / 16)) / 8;   // 32 blocks (K=32 or 68, Npad=32)
    const int blk6880 = (128 * (80 / 16)) / 8;   // 80 blocks (K=68, Npad=80)

    k_init<<<NODES / 8, 256, 0, stream>>>(V0, Yr, Yi, v, th, m, rs, sumA, loss);

    for (int k = 0; k < KITER; ++k) {
        float wk = powf(0.96f, (float)(KITER - 1 - k));

        k_powerflow<<<NODES / 8, 256, 0, stream>>>(Yr, Yi, bus_type, P_set, Q_set,
                                                   v, th, dP, dQ, loss, wk);

        // base = m @ edge_w1[0:32,:] + edge_b1   (2048 x 32, K=32)
        k_wmma_gemm<32, 32><<<blk3232, 256, 0, stream>>>(
            m, edge_w1, edge_b1, base, NODES, DD, DD, DD, DD, 0);

        k_edge<<<NODES, 256, 0, stream>>>(Yr, Yi, edge_w1, edge_w2, edge_b2,
                                          base, rs, sumA, m, v, th, dP, dQ,
                                          Mmsg, feats);

        for (int p = 0; p < 3; ++p) {
            const int od = (p == 2) ? DD : 1;
            const float* w1 = W[p][0] + (size_t)k * IND * IND;
            const float* b1 = W[p][1] + (size_t)k * IND;
            const float* w2 = W[p][2] + (size_t)k * IND * IND;
            const float* b2 = W[p][3] + (size_t)k * IND;
            const float* w3 = W[p][4] + (size_t)k * IND * od;
            const float* b3 = W[p][5] + (size_t)k * od;
            // h1 = leaky(feats @ w1 + b1); h2 = leaky(h1 @ w2 + b2)
            k_wmma_gemm<IND, 80><<<blk6880, 256, 0, stream>>>(
                feats, w1, b1, h1, NODES, IND, LDF, IND, LDF, 1);
            k_wmma_gemm<IND, 80><<<blk6880, 256, 0, stream>>>(
                h1, w2, b2, h2, NODES, IND, LDF, IND, LDF, 1);
            if (p != 2) {
                k_outdot<<<NODES / 8, 256, 0, stream>>>(h2, w3, b3, (p == 0) ? dth : dv);
            } else {
                // dm = tanh(h2 @ w3 + b3)   (2048 x 32, K=68)
                k_wmma_gemm<IND, 32><<<blk3232, 256, 0, stream>>>(
                    h2, w3, b3, dm, NODES, DD, LDF, DD, DD, 2);
            }
        }

        k_update<<<(NODES * DD + 255) / 256, 256, 0, stream>>>(bus_type, dth, dv, dm,
                                                               v, th, m);
    }

    k_final<<<(NODES + 1 + 255) / 256, 256, 0, stream>>>(v, th, loss, (float*)d_out);
}